// GATLayer_10849087390523
// MI455X (gfx1250) — compile-verified
//
#include <hip/hip_runtime.h>
#include <hip/hip_bf16.h>
#include <cmath>

// out = GELU_exact(x @ W^T + b)  -- GAT attention path is dead code, so
// adj (d_in[1], 256 MB) and a (d_in[4]) are never read.
//
// fp32 WMMA (V_WMMA_F32_16X16X4_F32), exact fp32 accumulation.
// Each wave: 16x128 output tile (8 n-tiles). 8 waves/block -> 128x128 tile.
//
// K-slot permutation: one b128 per lane at (base + 4*half) provides two WMMA
// fragments -- regs (r0,r1) cover K slots {k,k+1,k+4,k+5}, regs (r2,r3) cover
// {k+2,k+3,k+6,k+7} -- identically for A and B, so the sum is exact.
//
// Ping-pong double buffer: buffers alternate in place (no acc copies);
// per-lane stream pointers bumped once per 2 chunks, in-iteration loads use
// immediate offsets. All streams are explicit addrspace(1) so they lower to
// global_load_b128 (not flat_load: flat ticks DScnt and ties up the LDS path).

typedef float v2f __attribute__((ext_vector_type(2)));
typedef float v4f __attribute__((ext_vector_type(4)));
typedef float v8f __attribute__((ext_vector_type(8)));

typedef __attribute__((address_space(1))) const float* gcfp;
typedef __attribute__((address_space(1))) const v4f*   gcv4p;
typedef __attribute__((address_space(1))) float*       gfp;

#define NROWS 8192
#define D     512     // D_IN == D_OUT
#define TILE_M 128    // rows per block  (8 waves x 16)
#define TILE_N 128    // cols per block  (8 WMMA n-tiles per wave)

__device__ __forceinline__ float gelu_exact(float h) {
    return 0.5f * h * (1.0f + erff(h * 0.70710678118654752440f));
}

__device__ __forceinline__ v4f ldg4(gcfp p) {
    return *(gcv4p)p;
}

__device__ __forceinline__ void mma_chunk(v8f acc[8], v4f a, const v4f b[8]) {
    const v2f alo = __builtin_shufflevector(a, a, 0, 1);
    const v2f ahi = __builtin_shufflevector(a, a, 2, 3);
#pragma unroll
    for (int t = 0; t < 8; ++t) {
        const v2f blo = __builtin_shufflevector(b[t], b[t], 0, 1);
        acc[t] = __builtin_amdgcn_wmma_f32_16x16x4_f32(
            false, alo, false, blo, (short)0, acc[t], false, false);
    }
#pragma unroll
    for (int t = 0; t < 8; ++t) {
        const v2f bhi = __builtin_shufflevector(b[t], b[t], 2, 3);
        acc[t] = __builtin_amdgcn_wmma_f32_16x16x4_f32(
            false, ahi, false, bhi, (short)0, acc[t], false, false);
    }
}

__global__ __launch_bounds__(256) void gat_linear_gelu_f32wmma(
    const float* __restrict__ x,     // [8192, 512]
    const float* __restrict__ W,     // [512, 512] row-major [out, in]
    const float* __restrict__ bias,  // [512]
    float* __restrict__ out)         // [8192, 512]
{
    const int lane = threadIdx.x & 31;
    const int wave = threadIdx.x >> 5;
    const int half = lane >> 4;      // 0: lanes 0-15, 1: lanes 16-31
    const int l15  = lane & 15;

    const int mbase = blockIdx.x * TILE_M + wave * 16;
    const int nbase = blockIdx.y * TILE_N;

    // persistent per-lane global-addrspace stream pointers (1 A + 8 B)
    gcfp px = (gcfp)(x + (size_t)(mbase + l15) * D + 4 * half);
    gcfp pw[8];
#pragma unroll
    for (int t = 0; t < 8; ++t)
        pw[t] = (gcfp)(W + (size_t)(nbase + t * 16 + l15) * D + 4 * half);

    v8f acc[8];
#pragma unroll
    for (int t = 0; t < 8; ++t) acc[t] = (v8f)(0.0f);

    // ---- preload chunks 0 and 1 (8 K each) ----
    v4f a0 = ldg4(px);
    v4f a1 = ldg4(px + 8);
    v4f b0[8], b1[8];
#pragma unroll
    for (int t = 0; t < 8; ++t) b0[t] = ldg4(pw[t]);
#pragma unroll
    for (int t = 0; t < 8; ++t) b1[t] = ldg4(pw[t] + 8);

    // ---- ping-pong main loop: consume 2 chunks, load 2 chunks ----
#pragma unroll 1
    for (int it = 0; it < (D / 16) - 1; ++it) {   // 31 iterations
        mma_chunk(acc, a0, b0);                   // consume chunk 2it
        a0 = ldg4(px + 16);                       // load chunk 2it+2 (imm offs)
#pragma unroll
        for (int t = 0; t < 8; ++t) b0[t] = ldg4(pw[t] + 16);

        mma_chunk(acc, a1, b1);                   // consume chunk 2it+1
        a1 = ldg4(px + 24);                       // load chunk 2it+3 (imm offs)
#pragma unroll
        for (int t = 0; t < 8; ++t) b1[t] = ldg4(pw[t] + 24);

        px += 16;                                 // one bump per stream / 2 chunks
#pragma unroll
        for (int t = 0; t < 8; ++t) pw[t] += 16;
    }
    mma_chunk(acc, a0, b0);                       // chunks 62, 63
    mma_chunk(acc, a1, b1);

    // ---- epilogue: bias + exact GELU + non-temporal global store ----
    // C/D layout: acc[t][r] = (row mbase + 8*half + r, col nbase + 16*t + l15)
#pragma unroll
    for (int t = 0; t < 8; ++t) {
        const int n  = nbase + t * 16 + l15;
        const float bn = *((gcfp)(bias + n));
#pragma unroll
        for (int r = 0; r < 8; ++r) {
            const int row = mbase + half * 8 + r;
            __builtin_nontemporal_store(gelu_exact(acc[t][r] + bn),
                                        (gfp)(out + (size_t)row * D + n));
        }
    }
}

extern "C" void kernel_launch(void* const* d_in, const int* in_sizes, int n_in,
                              void* d_out, int out_size, void* d_ws, size_t ws_size,
                              hipStream_t stream) {
    const float* x  = (const float*)d_in[0];   // [8192,512]
    // d_in[1] = adj (int32 [8192,8192]) -- dead code, never read
    const float* W  = (const float*)d_in[2];   // [512,512]
    const float* b  = (const float*)d_in[3];   // [512]
    // d_in[4] = a -- dead code, never read
    float* out = (float*)d_out;                // [8192,512] fp32

    dim3 grid(NROWS / TILE_M, D / TILE_N);     // (64, 4)
    dim3 block(256);                           // 8 waves (wave32)
    gat_linear_gelu_f32wmma<<<grid, block, 0, stream>>>(x, W, b, out);
}